// Net_39943195853539
// MI455X (gfx1250) — compile-verified
//
#include <hip/hip_runtime.h>
#include <hip/hip_bf16.h>

typedef __attribute__((ext_vector_type(2))) float v2f;
typedef __attribute__((ext_vector_type(8))) float v8f;

#define KK 5
#define K2 25
#define NMAT 26            // 25 spline kernels + root matrix in slot 25
#define YSTRIDE (NMAT*16)  // 416 floats per node in Y

// ---------------------------------------------------------------- utilities
__global__ void zero_f32(float* __restrict__ p, long n) {
  long i = (long)blockIdx.x * blockDim.x + threadIdx.x;
  long stride = (long)gridDim.x * blockDim.x;
  for (; i < n; i += stride) p[i] = 0.0f;
}

// degree-1 open B-spline basis in 2D: 4 (basis, flat-kernel-index) pairs
__device__ __forceinline__ void spline4(float a0, float a1,
                                        float (&bas)[4], int (&wi)[4]) {
  float v0 = a0 * (float)(KK - 1), v1 = a1 * (float)(KK - 1);
  float l0 = floorf(v0), l1 = floorf(v1);
  float f0 = v0 - l0, f1 = v1 - l1;
  int i0 = (int)l0, i1 = (int)l1;
  int id0[2], id1[2];
  id0[0] = i0 < 0 ? 0 : (i0 > KK - 1 ? KK - 1 : i0);
  id0[1] = (i0 + 1) < 0 ? 0 : ((i0 + 1) > KK - 1 ? KK - 1 : i0 + 1);
  id1[0] = i1 < 0 ? 0 : (i1 > KK - 1 ? KK - 1 : i1);
  id1[1] = (i1 + 1) < 0 ? 0 : ((i1 + 1) > KK - 1 ? KK - 1 : i1 + 1);
  float g0[2] = {1.0f - f0, f0};
  float g1[2] = {1.0f - f1, f1};
#pragma unroll
  for (int t = 0; t < 4; ++t) {
    bas[t] = g0[t & 1] * g1[t >> 1];
    wi[t]  = id0[t & 1] * KK + id1[t >> 1];
  }
}

// -------------------------------------------------- layer-1 batched GEMM (WMMA)
// Y[n, k, j] = sum_c X[n,c] * Wk[c,j]   (Wk = W1[k] for k<25, root1 for k==25)
// X:[N,64] f32, W:[25,64,16], root:[64,16], Y:[N,26,16]
__global__ __launch_bounds__(256) void gemm64x16_wmma(
    const float* __restrict__ X, const float* __restrict__ W,
    const float* __restrict__ root, float* __restrict__ Y, int N) {
  const int lane = threadIdx.x & 31;
  const int wave = threadIdx.x >> 5;
  const int m  = lane & 15;   // A-matrix row
  const int hl = lane >> 4;   // half-wave selector
  const int tile = blockIdx.x;
  const int row = tile * 16 + m;
  const bool fullTile = (tile * 16 + 16) <= N;   // uniform across block

  // A fragments for 16 K-steps of 4: lane holds X[row, 4s+2hl .. +1]
  v2f a[16];
  if (fullTile || row < N) {
    const float* p = X + (long)row * 64 + 2 * hl;
#pragma unroll
    for (int s = 0; s < 16; ++s) a[s] = *(const v2f*)(p + 4 * s);
  } else {
#pragma unroll
    for (int s = 0; s < 16; ++s) a[s] = (v2f)0.0f;
  }

  const int n = lane & 15;    // B/D column
  for (int k = wave; k < NMAT; k += 8) {
    const float* Wk = (k < K2) ? (W + (long)k * 64 * 16) : root;
    v8f acc = (v8f)0.0f;
#pragma unroll
    for (int s = 0; s < 16; ++s) {
      const int c = 4 * s + 2 * hl;
      v2f b;
      b.x = Wk[c * 16 + n];
      b.y = Wk[(c + 1) * 16 + n];
      acc = __builtin_amdgcn_wmma_f32_16x16x4_f32(
          false, a[s], false, b, (short)0, acc, false, false);
    }
    float* yo = Y + (long)(tile * 16) * YSTRIDE + k * 16 + n;
    if (fullTile) {           // uniform scalar branch: no exec-mask churn
#pragma unroll
      for (int r = 0; r < 8; ++r)
        yo[(long)(r + 8 * hl) * YSTRIDE] = acc[r];
    } else {
#pragma unroll
      for (int r = 0; r < 8; ++r) {
        const int orow = tile * 16 + r + 8 * hl;
        if (orow < N) Y[(long)orow * YSTRIDE + k * 16 + n] = acc[r];
      }
    }
  }
}

// -------------------------------------------------- layer-2 batched GEMM (WMMA)
// Y[n, k, j] = sum_{c<16} H[n,c] * Wk[c,j], j<7 (N-dim padded to 16 with zeros)
__global__ __launch_bounds__(256) void gemm16x7_wmma(
    const float* __restrict__ H, const float* __restrict__ W,
    const float* __restrict__ root, float* __restrict__ Y, int N) {
  const int lane = threadIdx.x & 31;
  const int wave = threadIdx.x >> 5;
  const int m  = lane & 15;
  const int hl = lane >> 4;
  const int tile = blockIdx.x;
  const int row = tile * 16 + m;
  const bool fullTile = (tile * 16 + 16) <= N;

  v2f a[4];
  if (fullTile || row < N) {
    const float* p = H + (long)row * 16 + 2 * hl;
#pragma unroll
    for (int s = 0; s < 4; ++s) a[s] = *(const v2f*)(p + 4 * s);
  } else {
#pragma unroll
    for (int s = 0; s < 4; ++s) a[s] = (v2f)0.0f;
  }

  const int n = lane & 15;
  for (int k = wave; k < NMAT; k += 8) {
    const float* Wk = (k < K2) ? (W + (long)k * 16 * 7) : root;
    v8f acc = (v8f)0.0f;
#pragma unroll
    for (int s = 0; s < 4; ++s) {
      const int c = 4 * s + 2 * hl;
      v2f b;
      b.x = (n < 7) ? Wk[c * 7 + n] : 0.0f;
      b.y = (n < 7) ? Wk[(c + 1) * 7 + n] : 0.0f;
      acc = __builtin_amdgcn_wmma_f32_16x16x4_f32(
          false, a[s], false, b, (short)0, acc, false, false);
    }
    float* yo = Y + (long)(tile * 16) * YSTRIDE + k * 16 + n;
    if (fullTile) {
#pragma unroll
      for (int r = 0; r < 8; ++r)
        yo[(long)(r + 8 * hl) * YSTRIDE] = acc[r];
    } else {
#pragma unroll
      for (int r = 0; r < 8; ++r) {
        const int orow = tile * 16 + r + 8 * hl;
        if (orow < N) Y[(long)orow * YSTRIDE + k * 16 + n] = acc[r];
      }
    }
  }
}

// ---------------------------------------------------------- edge passes
// 16 lanes per edge, one output channel each
__global__ void edge_pass1(const long long* __restrict__ ei,
                           const float* __restrict__ attr,
                           const float* __restrict__ Y,
                           float* __restrict__ agg, float* __restrict__ deg,
                           int E) {
  long g = (long)blockIdx.x * blockDim.x + threadIdx.x;
  int e = (int)(g >> 4);
  int c = (int)(g & 15);
  if (e >= E) return;
  int src = (int)ei[e];
  int dst = (int)ei[(long)E + e];
  float bas[4]; int wi[4];
  spline4(attr[2 * e], attr[2 * e + 1], bas, wi);
  const float* ys = Y + (long)src * YSTRIDE;
  float msg = 0.0f;
#pragma unroll
  for (int t = 0; t < 4; ++t) msg += bas[t] * ys[wi[t] * 16 + c];
  atomicAdd(agg + (long)dst * 16 + c, msg);
  if (c == 0) atomicAdd(deg + dst, 1.0f);
}

// 8 lanes per edge, channels 0..6
__global__ void edge_pass2(const long long* __restrict__ ei,
                           const float* __restrict__ attr,
                           const float* __restrict__ Y,
                           float* __restrict__ agg, int E) {
  long g = (long)blockIdx.x * blockDim.x + threadIdx.x;
  int e = (int)(g >> 3);
  int c = (int)(g & 7);
  if (e >= E || c >= 7) return;
  int src = (int)ei[e];
  int dst = (int)ei[(long)E + e];
  float bas[4]; int wi[4];
  spline4(attr[2 * e], attr[2 * e + 1], bas, wi);
  const float* ys = Y + (long)src * YSTRIDE;
  float msg = 0.0f;
#pragma unroll
  for (int t = 0; t < 4; ++t) msg += bas[t] * ys[wi[t] * 16 + c];
  atomicAdd(agg + (long)dst * 8 + c, msg);
}

// ---------------------------------------------------------- node passes
__global__ void node_pass1(const float* __restrict__ agg,
                           const float* __restrict__ deg,
                           const float* __restrict__ Y,
                           const float* __restrict__ bias,
                           float* __restrict__ H, int N) {
  long g = (long)blockIdx.x * blockDim.x + threadIdx.x;
  int nn = (int)(g >> 4);
  int c = (int)(g & 15);
  if (nn >= N) return;
  float d = deg[nn];
  d = d < 1.0f ? 1.0f : d;
  float v = agg[(long)nn * 16 + c] / d + Y[(long)nn * YSTRIDE + K2 * 16 + c] + bias[c];
  H[(long)nn * 16 + c] = v > 0.0f ? v : 0.0f;
}

__global__ void node_pass2(const float* __restrict__ agg,
                           const float* __restrict__ deg,
                           const float* __restrict__ Y,
                           const float* __restrict__ bias,
                           float* __restrict__ out, int N) {
  int nn = blockIdx.x * blockDim.x + threadIdx.x;
  if (nn >= N) return;
  float d = deg[nn];
  d = d < 1.0f ? 1.0f : d;
  float o[7];
#pragma unroll
  for (int j = 0; j < 7; ++j)
    o[j] = agg[(long)nn * 8 + j] / d + Y[(long)nn * YSTRIDE + K2 * 16 + j] + bias[j];
  float mx = o[0];
#pragma unroll
  for (int j = 1; j < 7; ++j) mx = o[j] > mx ? o[j] : mx;
  float s = 0.0f;
#pragma unroll
  for (int j = 0; j < 7; ++j) s += __expf(o[j] - mx);
  float ls = __logf(s);
#pragma unroll
  for (int j = 0; j < 7; ++j) out[(long)nn * 7 + j] = o[j] - mx - ls;
}

// ---------------------------------------------------------------- launcher
extern "C" void kernel_launch(void* const* d_in, const int* in_sizes, int n_in,
                              void* d_out, int out_size, void* d_ws, size_t ws_size,
                              hipStream_t stream) {
  const float*     x     = (const float*)d_in[0];
  const long long* ei    = (const long long*)d_in[1];  // reference dtype int64
  const float*     attr  = (const float*)d_in[2];
  const float*     W1    = (const float*)d_in[3];
  const float*     root1 = (const float*)d_in[4];
  const float*     bias1 = (const float*)d_in[5];
  const float*     W2    = (const float*)d_in[6];
  const float*     root2 = (const float*)d_in[7];
  const float*     bias2 = (const float*)d_in[8];
  float* out = (float*)d_out;

  const int N = in_sizes[0] / 64;
  const int E = in_sizes[2] / 2;

  // workspace carve-up (floats)
  float* Y    = (float*)d_ws;                 // N*416  (83.2 MB @ N=50k; L2-resident)
  float* agg1 = Y    + (long)N * YSTRIDE;     // N*16
  float* agg2 = agg1 + (long)N * 16;          // N*8
  float* deg  = agg2 + (long)N * 8;           // N
  float* H    = deg  + (long)N;               // N*16

  // zero agg1 | agg2 | deg (contiguous)
  zero_f32<<<1024, 256, 0, stream>>>(agg1, (long)N * 25);

  const int NT = (N + 15) / 16;
  gemm64x16_wmma<<<NT, 256, 0, stream>>>(x, W1, root1, Y, N);
  edge_pass1<<<(int)(((long)E * 16 + 255) / 256), 256, 0, stream>>>(ei, attr, Y, agg1, deg, E);
  node_pass1<<<(int)(((long)N * 16 + 255) / 256), 256, 0, stream>>>(agg1, deg, Y, bias1, H, N);
  gemm16x7_wmma<<<NT, 256, 0, stream>>>(H, W2, root2, Y, N);
  edge_pass2<<<(int)(((long)E * 8 + 255) / 256), 256, 0, stream>>>(ei, attr, Y, agg2, E);
  node_pass2<<<(N + 255) / 256, 256, 0, stream>>>(agg2, deg, Y, bias2, out, N);
}